// OverlapReward_65025804862187
// MI455X (gfx1250) — compile-verified
//
#include <hip/hip_runtime.h>
#include <stdint.h>
#include <math.h>

// Problem constants from the reference: B=4, A=64, T=80, pose=5 floats.
#define AGENTS 64
#define TSTEPS 80
#define ROW_TILES 4                     // split the 64x64 pair matrix into 4 x (16x64)
#define ROWS_PER_TILE (AGENTS / ROW_TILES)

__global__ __launch_bounds__(256)
void OverlapReward_overlap_sd_kernel(const float* __restrict__ traj,
                                     const int*   __restrict__ mask,
                                     float*       __restrict__ out)
{
    __shared__ float sPose[AGENTS * 5];          // staged via async global->LDS
    __shared__ float sCx[AGENTS][4];             // box corners (CCW)
    __shared__ float sCy[AGENTS][4];
    __shared__ float sDPx[AGENTS], sDPy[AGENTS]; // downmost edge dir of +corners
    __shared__ float sDNx[AGENTS], sDNy[AGENTS]; // downmost edge dir of -corners
    __shared__ int   sIP[AGENTS], sIN[AGENTS];   // downmost vertex indices
    __shared__ int   sM[AGENTS];                 // validity mask

    const int tid = threadIdx.x;
    const int rt  = blockIdx.x & (ROW_TILES - 1);    // which 16-row slice
    const int bt  = blockIdx.x >> 2;                 // (b,t) index
    const int b   = bt / TSTEPS;
    const int t   = bt - b * TSTEPS;

    // ---- Phase 1: stage 64 poses (320 floats) into LDS with CDNA5 async
    //      copies (tracked by ASYNCcnt). traj element (b,a,t,c) lives at
    //      ((b*A + a)*T + t)*5 + c.
    for (int f = tid; f < AGENTS * 5; f += 256) {
        const int a = f / 5;
        const int c = f - a * 5;
        const unsigned goff = (unsigned)((((b * AGENTS + a) * TSTEPS + t) * 5 + c) * 4u);
        // Low 32 bits of the generic pointer to a __shared__ object are the
        // workgroup-relative LDS byte offset (flat->LDS aperture mapping).
        const unsigned loff = (unsigned)(uintptr_t)(void*)(&sPose[f]);
        asm volatile("global_load_async_to_lds_b32 %0, %1, %2"
                     :: "v"(loff), "v"(goff), "s"(traj)
                     : "memory");
    }
    if (tid < AGENTS) {
        sM[tid] = mask[(b * AGENTS + tid) * TSTEPS + t];
    }
    asm volatile("s_wait_asynccnt 0" ::: "memory");
    __syncthreads();

    // ---- Phase 2: per-box corners + downmost-edge info (for +corners and
    //      for the negated corners used as box2 in the Minkowski sum).
    if (tid < AGENTS) {
        const float x   = sPose[tid * 5 + 0];
        const float y   = sPose[tid * 5 + 1];
        const float l   = sPose[tid * 5 + 2];
        const float w   = sPose[tid * 5 + 3];
        const float yaw = sPose[tid * 5 + 4];
        const float c = cosf(yaw), s = sinf(yaw);
        const float hl = 0.5f * l, hw = 0.5f * w;
        const float dx[4] = { hl, -hl, -hl,  hl };
        const float dy[4] = { hw,  hw, -hw, -hw };
        float cx[4], cy[4];
        #pragma unroll
        for (int k = 0; k < 4; ++k) {
            cx[k] = x + dx[k] * c - dy[k] * s;
            cy[k] = y + dx[k] * s + dy[k] * c;
            sCx[tid][k] = cx[k];
            sCy[tid][k] = cy[k];
        }
        // argmin over y, first occurrence (matches jnp.argmin)
        int ip = 0; float mp = cy[0];
        #pragma unroll
        for (int k = 1; k < 4; ++k) { if (cy[k] < mp) { mp = cy[k]; ip = k; } }
        {
            const int ipn = (ip + 1) & 3;
            const float ex = cx[ipn] - cx[ip];
            const float ey = cy[ipn] - cy[ip];
            const float n  = sqrtf(ex * ex + ey * ey);
            sDPx[tid] = ex / n;  sDPy[tid] = ey / n;
            sIP[tid]  = ip;
        }
        // same for negated corners (box2 = -corners_j)
        float nxv[4], nyv[4];
        #pragma unroll
        for (int k = 0; k < 4; ++k) { nxv[k] = -cx[k]; nyv[k] = -cy[k]; }
        int iq = 0; float mq = nyv[0];
        #pragma unroll
        for (int k = 1; k < 4; ++k) { if (nyv[k] < mq) { mq = nyv[k]; iq = k; } }
        {
            const int iqn = (iq + 1) & 3;
            const float fx = nxv[iqn] - nxv[iq];
            const float fy = nyv[iqn] - nyv[iq];
            const float n2 = sqrtf(fx * fx + fy * fy);
            sDNx[tid] = fx / n2;  sDNy[tid] = fy / n2;
            sIN[tid]  = iq;
        }
    }
    __syncthreads();

    // ---- Phase 3: this block's 16x64 pair slice; thread handles 4 pairs.
    const int po1[8] = {0, 0, 1, 1, 2, 2, 3, 3};
    const int po2[8] = {0, 1, 1, 2, 2, 3, 3, 0};

    for (int p = tid; p < ROWS_PER_TILE * AGENTS; p += 256) {
        const int i = rt * ROWS_PER_TILE + (p >> 6);
        const int j = p & 63;

        const float d1x = sDPx[i], d1y = sDPy[i];
        const float d2x = sDNx[j], d2y = sDNy[j];
        const bool cond = (d1x * d2y - d1y * d2x) >= 0.0f;
        const int i1 = sIP[i];
        const int i2 = sIN[j];

        // Minkowski sum vertices: corners_i[o1] + (-corners_j)[o2]
        float vx[8], vy[8];
        #pragma unroll
        for (int k = 0; k < 8; ++k) {
            const int s1 = cond ? po2[k] : po1[k];
            const int s2 = cond ? po1[k] : po2[k];
            const int o1 = (s1 + i1) & 3;
            const int o2 = (s2 + i2) & 3;
            vx[k] = sCx[i][o1] - sCx[j][o2];
            vy[k] = sCy[i][o1] - sCy[j][o2];
        }

        // signed distance of origin to the convex 8-gon
        float mind = INFINITY;
        #pragma unroll
        for (int k = 0; k < 8; ++k) {
            mind = fminf(mind, sqrtf(vx[k] * vx[k] + vy[k] * vy[k]));
        }
        bool inside = true;
        #pragma unroll
        for (int k = 0; k < 8; ++k) {
            const int kn = (k + 1) & 7;
            const float ex = vx[kn] - vx[k];
            const float ey = vy[kn] - vy[k];
            const float el = sqrtf(ex * ex + ey * ey);
            const float tux = ex / el, tuy = ey / el;
            // sperp = sum(-nu * (q - v)) with q=origin  ->  tux*vy - tuy*vx
            const float sperp = tux * vy[k] - tuy * vx[k];
            inside = inside && (sperp <= 0.0f);
            const float proj = -(tux * vx[k] + tuy * vy[k]) / el;
            if (proj >= 0.0f && proj <= 1.0f) {
                mind = fminf(mind, fabsf(sperp));
            }
        }
        const float sd  = inside ? -mind : mind;
        const float res = ((i != j) && (sM[i] != 0) && (sM[j] != 0)) ? sd : 1000.0f;

        // out shape (B, A, T, A): out[b][i][t][j]
        out[(((size_t)b * AGENTS + i) * TSTEPS + t) * AGENTS + j] = res;
    }
}

extern "C" void kernel_launch(void* const* d_in, const int* in_sizes, int n_in,
                              void* d_out, int out_size, void* d_ws, size_t ws_size,
                              hipStream_t stream) {
    (void)n_in; (void)d_ws; (void)ws_size; (void)out_size;
    const float* traj = (const float*)d_in[0];
    const int*   mask = (const int*)d_in[1];
    float*       out  = (float*)d_out;

    // in_sizes[0] = B*A*T*5 -> number of (b,t) pairs = B*T, 4 row-tiles each
    const int B = in_sizes[0] / (AGENTS * TSTEPS * 5);
    OverlapReward_overlap_sd_kernel<<<B * TSTEPS * ROW_TILES, 256, 0, stream>>>(traj, mask, out);
}